// FastRCNN_16286515987057
// MI455X (gfx1250) — compile-verified
//
#include <hip/hip_runtime.h>
#include <hip/hip_bf16.h>

// ---------------------------------------------------------------------------
// FastRCNN head for MI455X (gfx1250, wave32, WMMA).
//   1) roi_align : feature_map[16,16,2048] + 64 boxes -> X[64, 49*2048]
//   2) H1 = relu(X @ W1 + b1)     (K=100352, dominant: streams 411MB of W1)
//   3) H2 = relu(H1 @ W2 + b2)    (K=1024)
//   4) cls = softmax(H2 @ Wc + bc), bbox = H2 @ Wb + bb
// GEMMs use V_WMMA_F32_16X16X4_F32; K split across waves, merged with
// global f32 atomics into a bias-initialized accumulator.
// Weight (B) loads are non-temporal so the 411MB W1 stream does not evict
// the L2-resident activation matrix X (25.7MB, reused by all 64 N-tiles).
// ---------------------------------------------------------------------------

typedef __attribute__((ext_vector_type(2))) float v2f;
typedef __attribute__((ext_vector_type(8))) float v8f;

#define NH   1024          // hidden width (N of both big GEMMs)
#define K1   100352        // 7*7*2048
#define XELT (64 * K1)     // floats in X
#define HELT (64 * NH)     // floats in H1/H2

// ---------------- ROI align (crop + TF2 half-pixel bilinear resize) --------
__global__ void __launch_bounds__(256)
roi_align_kernel(const float* __restrict__ fm,      // [16,16,2048]
                 const float* __restrict__ props,   // [64,4] x1,y1,x2,y2
                 float* __restrict__ X)             // [64, 49*2048]
{
    const int cell = blockIdx.x;        // 0..64*49-1
    const int r  = cell / 49;
    const int sc = cell % 49;
    const int sy = sc / 7, sx = sc % 7;

    const float4 box = ((const float4*)props)[r];
    const float x1 = fmaxf(0.0f,  rintf(box.x));
    const float y1 = fmaxf(0.0f,  rintf(box.y));
    const float x2 = fminf(15.0f, rintf(box.z));
    const float y2 = fminf(15.0f, rintf(box.w));
    const float hh = y2 - y1;
    const float ww = x2 - x1;

    // half-pixel sample coords relative to crop, clip like jnp.clip (max then min)
    float iy = fminf(fmaxf(((float)sy + 0.5f) * hh * (1.0f / 7.0f) - 0.5f, 0.0f), hh - 1.0f);
    float ix = fminf(fmaxf(((float)sx + 0.5f) * ww * (1.0f / 7.0f) - 0.5f, 0.0f), ww - 1.0f);
    const float fy = iy - floorf(iy);
    const float fx = ix - floorf(ix);
    const int y0i = (int)(y1 + floorf(iy));
    const int y1i = min(y0i + 1, (int)(y1 + hh - 1.0f));
    const int x0i = (int)(x1 + floorf(ix));
    const int x1i = min(x0i + 1, (int)(x1 + ww - 1.0f));

    const float* p00 = fm + ((size_t)y0i * 16 + x0i) * 2048;
    const float* p01 = fm + ((size_t)y0i * 16 + x1i) * 2048;
    const float* p10 = fm + ((size_t)y1i * 16 + x0i) * 2048;
    const float* p11 = fm + ((size_t)y1i * 16 + x1i) * 2048;
    float* dst = X + ((size_t)r * 49 + sc) * 2048;

    const float w00 = (1.0f - fy) * (1.0f - fx);
    const float w01 = (1.0f - fy) * fx;
    const float w10 = fy * (1.0f - fx);
    const float w11 = fy * fx;

    for (int c = threadIdx.x; c < 2048; c += 256)
        dst[c] = p00[c] * w00 + p01[c] * w01 + p10[c] * w10 + p11[c] * w11;
}

// ---------------- C[m,n] = bias[n] (accumulator init) ----------------------
__global__ void __launch_bounds__(256)
init_bias_kernel(float* __restrict__ C, const float* __restrict__ bias)
{
    const int idx = blockIdx.x * 256 + threadIdx.x;   // 64*1024 elements
    if (idx < HELT) C[idx] = bias[idx & (NH - 1)];
}

// ---------------- in-place ReLU --------------------------------------------
__global__ void __launch_bounds__(256)
relu_kernel(float* __restrict__ X, int n)
{
    const int idx = blockIdx.x * 256 + threadIdx.x;
    if (idx < n) X[idx] = fmaxf(X[idx], 0.0f);
}

// ---------------- K-split skinny GEMM via V_WMMA_F32_16X16X4_F32 -----------
// C[64, 1024] += A[64, K] @ B[K, 1024]
// grid.x = 64 N-tiles (16 cols), grid.y * 8 waves = K chunks of chunkLen.
// B (weights) loaded non-temporal: streamed once, must not evict A from L2.
__global__ void __launch_bounds__(256)
gemm_wmma_f32(const float* __restrict__ A, const float* __restrict__ B,
              float* __restrict__ Cacc, int K, int chunkLen)
{
    const int lane   = threadIdx.x & 31;
    const int wave   = threadIdx.x >> 5;       // 0..7
    const int laneHi = lane >> 4;              // 0/1: half-wave
    const int lm     = lane & 15;
    const int nBase  = blockIdx.x * 16;
    const int chunk  = blockIdx.y * 8 + wave;
    const int k0     = chunk * chunkLen;
    const int kOff   = laneHi * 2;             // K offset held by this half-wave

    v8f acc0 = {}, acc1 = {}, acc2 = {}, acc3 = {};

    // B 4x16: vgpr0 = B[k+kOff, nBase+lm], vgpr1 = B[k+kOff+1, nBase+lm]
    const float* pB0 = B + (size_t)(k0 + kOff) * NH + nBase + lm;
    const float* pB1 = pB0 + NH;
    // A 16x4 per M-tile t: lane holds A[16t+lm, k+kOff .. k+kOff+1] (contig pair)
    const float* pA0 = A + (size_t)(0 * 16 + lm) * K + k0 + kOff;
    const float* pA1 = A + (size_t)(1 * 16 + lm) * K + k0 + kOff;
    const float* pA2 = A + (size_t)(2 * 16 + lm) * K + k0 + kOff;
    const float* pA3 = A + (size_t)(3 * 16 + lm) * K + k0 + kOff;

    const int iters = chunkLen >> 2;           // always a multiple of 4 here
#pragma unroll 4
    for (int i = 0; i < iters; ++i) {
        v2f b;
        b.x = __builtin_nontemporal_load(pB0);  // TH=NT: protect L2-resident A
        b.y = __builtin_nontemporal_load(pB1);
        v2f a0 = *(const v2f*)pA0;
        v2f a1 = *(const v2f*)pA1;
        v2f a2 = *(const v2f*)pA2;
        v2f a3 = *(const v2f*)pA3;
        acc0 = __builtin_amdgcn_wmma_f32_16x16x4_f32(false, a0, false, b, (short)0, acc0, false, false);
        acc1 = __builtin_amdgcn_wmma_f32_16x16x4_f32(false, a1, false, b, (short)0, acc1, false, false);
        acc2 = __builtin_amdgcn_wmma_f32_16x16x4_f32(false, a2, false, b, (short)0, acc2, false, false);
        acc3 = __builtin_amdgcn_wmma_f32_16x16x4_f32(false, a3, false, b, (short)0, acc3, false, false);
        pB0 += 4 * NH;  pB1 += 4 * NH;
        pA0 += 4;       pA1 += 4;       pA2 += 4;       pA3 += 4;
    }

    // C/D layout: vgpr j -> row (j + 8*laneHi), col lm (within 16x16 tile)
    float* base = Cacc + (size_t)(8 * laneHi) * NH + nBase + lm;
#pragma unroll
    for (int j = 0; j < 8; ++j) {
        atomicAdd(base + (size_t)(0 * 16 + j) * NH, acc0[j]);
        atomicAdd(base + (size_t)(1 * 16 + j) * NH, acc1[j]);
        atomicAdd(base + (size_t)(2 * 16 + j) * NH, acc2[j]);
        atomicAdd(base + (size_t)(3 * 16 + j) * NH, acc3[j]);
    }
}

// ---------------- classification / bbox heads + softmax --------------------
__global__ void __launch_bounds__(192)
heads_kernel(const float* __restrict__ H2,
             const float* __restrict__ Wc, const float* __restrict__ bc,
             const float* __restrict__ Wb, const float* __restrict__ bb,
             float* __restrict__ out)   // [64*2 cls | 64*4 bbox]
{
    __shared__ float logits[2];
    const int m    = blockIdx.x;
    const int w    = threadIdx.x >> 5;     // 0..5 -> output column
    const int lane = threadIdx.x & 31;
    const float* h = H2 + (size_t)m * NH;

    float s = 0.0f;
    if (w < 2) {
        for (int i = lane; i < NH; i += 32) s += h[i] * Wc[i * 2 + w];
    } else {
        const int col = w - 2;
        for (int i = lane; i < NH; i += 32) s += h[i] * Wb[i * 4 + col];
    }
#pragma unroll
    for (int off = 16; off > 0; off >>= 1) s += __shfl_down(s, off, 32);

    if (lane == 0) {
        if (w < 2) logits[w] = s + bc[w];
        else       out[128 + m * 4 + (w - 2)] = s + bb[w - 2];
    }
    __syncthreads();
    if (threadIdx.x == 0) {
        const float l0 = logits[0], l1 = logits[1];
        const float mx = fmaxf(l0, l1);
        const float e0 = expf(l0 - mx), e1 = expf(l1 - mx);
        const float inv = 1.0f / (e0 + e1);
        out[m * 2 + 0] = e0 * inv;
        out[m * 2 + 1] = e1 * inv;
    }
}

// ---------------------------------------------------------------------------
extern "C" void kernel_launch(void* const* d_in, const int* in_sizes, int n_in,
                              void* d_out, int out_size, void* d_ws, size_t ws_size,
                              hipStream_t stream)
{
    const float* fm    = (const float*)d_in[0];
    const float* props = (const float*)d_in[1];
    const float* W1    = (const float*)d_in[2];
    const float* b1    = (const float*)d_in[3];
    const float* W2    = (const float*)d_in[4];
    const float* b2    = (const float*)d_in[5];
    const float* Wc    = (const float*)d_in[6];
    const float* bc    = (const float*)d_in[7];
    const float* Wb    = (const float*)d_in[8];
    const float* bb    = (const float*)d_in[9];
    float* out = (float*)d_out;

    float* X  = (float*)d_ws;          // 64*100352 f32  (~25.7 MB, L2-resident)
    float* H1 = X + XELT;              // 64*1024
    float* H2 = H1 + HELT;             // 64*1024

    // 1) ROI align -> X
    roi_align_kernel<<<64 * 49, 256, 0, stream>>>(fm, props, X);

    // 2) H1 = relu(X @ W1 + b1): K=100352 split into 64 chunks of 1568
    init_bias_kernel<<<(HELT + 255) / 256, 256, 0, stream>>>(H1, b1);
    gemm_wmma_f32<<<dim3(64, 8), 256, 0, stream>>>(X, W1, H1, K1, 1568);
    relu_kernel<<<(HELT + 255) / 256, 256, 0, stream>>>(H1, HELT);

    // 3) H2 = relu(H1 @ W2 + b2): K=1024 split into 16 chunks of 64
    init_bias_kernel<<<(HELT + 255) / 256, 256, 0, stream>>>(H2, b2);
    gemm_wmma_f32<<<dim3(64, 2), 256, 0, stream>>>(H1, W2, H2, NH, 64);
    relu_kernel<<<(HELT + 255) / 256, 256, 0, stream>>>(H2, HELT);

    // 4) heads: softmax(H2 @ Wc + bc), H2 @ Wb + bb
    heads_kernel<<<64, 192, 0, stream>>>(H2, Wc, bc, Wb, bb, out);
}